// unit_spd_36661840838900
// MI455X (gfx1250) — compile-verified
//
#include <hip/hip_runtime.h>
#include <math.h>

typedef float v2f __attribute__((ext_vector_type(2)));
typedef float v8f __attribute__((ext_vector_type(8)));

#define N0 289
#define N1 100
#define N2 25

// D = A(16x4) * B(4x16) + C, full fp32 (keeps eigen/log precision downstream)
static __device__ __forceinline__ v8f wmma4(v2f a, v2f b, v8f c) {
  return __builtin_amdgcn_wmma_f32_16x16x4_f32(false, a, false, b, (short)0, c,
                                               false, false);
}

static __device__ __forceinline__ v8f vzero8() {
  v8f z;
#pragma unroll
  for (int r = 0; r < 8; ++r) z[r] = 0.0f;
  return z;
}

static __device__ __forceinline__ int imin(int a, int b) { return a < b ? a : b; }

// ---------------------------------------------------------------------------
// Kernel A: per batch   Z = sinh( W2 * sinh( W1 * X * W1^T ) * W2^T )
// One workgroup (8 wave32) per batch. X streamed once from HBM in 64-col
// panels; panel T, Y1, T2 live in LDS.
//
// Guarding strategy (keeps hot loops free of exec-mask/predication):
//  * M/N (non-contracted) out-of-range indices are CLAMPED (in-bounds loads,
//    duplicated data). Garbage lands only in pad rows/cols of intermediates.
//  * Pad rows/cols of Y1 are explicitly zeroed at deposit time, so every
//    later unguarded consumer sees exact zeros (sinh(0)=0; no Inf/NaN risk).
//  * K (contracted) boundaries get zeros via clamp + v_cndmask, only in the
//    single tail step (stage 1) / last panel (stage 2).
//
// WMMA f32 16x16x4 fragments (wave32):
//   A: lane m = l%16 holds (m, k0+2*hi), (m, k0+2*hi+1)
//   B: lane n = l%16 holds (k0+2*hi, n), (k0+2*hi+1, n)
//   C/D: vgpr r, lane -> (M = r + 8*hi, N = l%16)
// ---------------------------------------------------------------------------

template <bool GUARD_K>
static __device__ __forceinline__ v8f stage2_panel(
    v8f acc, const float (*Tb)[65], const float* __restrict__ pBn, int mrow,
    int khi, int nks, int jg0) {
  for (int kk = 0; kk < nks; ++kk) {
    const int jl = kk * 4 + khi;
    v2f a, bb;
    a[0] = Tb[mrow][jl];
    a[1] = Tb[mrow][jl + 1];
    if (GUARD_K) {
      const int jg = jg0 + jl;
      const float f0 = pBn[imin(jg, N0 - 1)];
      const float f1 = pBn[imin(jg + 1, N0 - 1)];
      bb[0] = (jg < N0) ? f0 : 0.0f;
      bb[1] = (jg + 1 < N0) ? f1 : 0.0f;
    } else {
      bb[0] = pBn[jg0 + jl];
      bb[1] = pBn[jg0 + jl + 1];
    }
    acc = wmma4(a, bb, acc);
  }
  return acc;
}

__global__ __launch_bounds__(256) void spd_bimap_kernel(
    const float* __restrict__ X, const float* __restrict__ W1,
    const float* __restrict__ W2, float* __restrict__ Z) {
  __shared__ float Tb[112][65];   // stage-1 panel (stride 65: conflict-free)
  __shared__ float Y1[112][113];  // sinh(W1 X W1^T), pad region exactly zero
  __shared__ float T2[32][113];   // W2 * Y1

  const int b = blockIdx.x;
  const float* Xb = X + (size_t)b * (N0 * N0);
  const int tid = threadIdx.x;
  const int w = tid >> 5;  // wave id 0..7
  const int l = tid & 31;
  const int lm = l & 15;
  const int hi = l >> 4;  // 0 | 1
  const int khi = 2 * hi;

  // stage-2 accumulators: 7x7 = 49 Y tiles round-robined over 8 waves
  v8f accY[7];
#pragma unroll
  for (int i = 0; i < 7; ++i) accY[i] = vzero8();

  for (int cb = 0; cb < 5; ++cb) {  // 5 panels of <=4 column tiles (19 total)
    const int jt0 = cb * 4;
    const int ntb = imin(4, 19 - jt0);

    // ---- stage 1: Tb = W1 * X[:, panel]  (clamped rows/cols, clean K loop)
    const int tot = 7 * ntb;
    for (int t = w; t < tot; t += 8) {
      const int mt = t % 7;
      const int jl = t / 7;
      const int m = imin(mt * 16 + lm, N1 - 1);          // clamped W1 row
      const int j = imin((jt0 + jl) * 16 + lm, N0 - 1);  // clamped X col
      const float* __restrict__ pA = W1 + m * N0;
      const float* __restrict__ pB = Xb + j;
      v8f acc = vzero8();
#pragma unroll 4
      for (int ks = 0; ks < 72; ++ks) {  // k, k+1 <= 287: no guards at all
        const int k = ks * 4 + khi;
        v2f a, bb;
        a[0] = pA[k];
        a[1] = pA[k + 1];
        bb[0] = pB[k * N0];
        bb[1] = pB[(k + 1) * N0];
        acc = wmma4(a, bb, acc);
      }
      {  // tail step ks=72: only (hi==0, first element, k=288) is real
        v2f a, bb;
        const float a0 = pA[N0 - 1];
        const float b0 = pB[(N0 - 1) * N0];
        a[0] = hi ? 0.0f : a0;
        a[1] = 0.0f;
        bb[0] = hi ? 0.0f : b0;
        bb[1] = 0.0f;
        acc = wmma4(a, bb, acc);
      }
      const int jc = jl * 16 + lm;
#pragma unroll
      for (int r = 0; r < 8; ++r) Tb[mt * 16 + r + 8 * hi][jc] = acc[r];
    }
    __syncthreads();

    // ---- stage 2 (partial): Y += Tb * W1^T over this panel ----
    const int nks = ntb * 4;
    const int jg0 = cb * 64;
#pragma unroll
    for (int i = 0; i < 7; ++i) {
      const int t = w + 8 * i;
      if (t < 49) {  // wave-uniform guard
        const int ot = t % 7;
        const int pt = t / 7;
        const int mrow = ot * 16 + lm;
        const int n = imin(pt * 16 + lm, N1 - 1);  // clamped W1 row (=Y col)
        const float* __restrict__ pBn = W1 + n * N0;
        accY[i] = (cb < 4)
                      ? stage2_panel<false>(accY[i], Tb, pBn, mrow, khi, nks, jg0)
                      : stage2_panel<true>(accY[i], Tb, pBn, mrow, khi, nks, jg0);
      }
    }
    __syncthreads();
  }

  // ---- sinh; deposit Y1 with pad region forced to exact zero ----
#pragma unroll
  for (int i = 0; i < 7; ++i) {
    const int t = w + 8 * i;
    if (t < 49) {
      const int ot = t % 7;
      const int pt = t / 7;
      const int p = pt * 16 + lm;
#pragma unroll
      for (int r = 0; r < 8; ++r) {
        const int o = ot * 16 + r + 8 * hi;
        const float v = sinhf(accY[i][r]);
        Y1[o][p] = (o < N1 && p < N1) ? v : 0.0f;
      }
    }
  }
  __syncthreads();

  // ---- bimap2 stage 1: T2 = W2 * Y1  (2 x 7 tiles, K = 100 = 25 steps) ----
  for (int t = w; t < 14; t += 8) {
    const int mt = t & 1;
    const int jt = t >> 1;
    const int m = imin(mt * 16 + lm, N2 - 1);  // clamped W2 row
    const int n = jt * 16 + lm;                // Y1 col (pad cols are zero)
    const float* __restrict__ pA = W2 + m * N1;
    v8f acc = vzero8();
#pragma unroll 5
    for (int ks = 0; ks < 25; ++ks) {
      const int k = ks * 4 + khi;
      v2f a, bb;
      a[0] = pA[k];
      a[1] = pA[k + 1];
      bb[0] = Y1[k][n];
      bb[1] = Y1[k + 1][n];
      acc = wmma4(a, bb, acc);
    }
#pragma unroll
    for (int r = 0; r < 8; ++r) T2[mt * 16 + r + 8 * hi][jt * 16 + lm] = acc[r];
  }
  __syncthreads();

  // ---- bimap2 stage 2: Z = sinh(T2 * W2^T)  (2 x 2 tiles, K = 100) ----
  if (w < 4) {
    const int ot = w & 1;
    const int pt = w >> 1;
    const int m = ot * 16 + lm;                 // T2 row (rows >=25: garbage,
    const int n = pt * 16 + lm;                 //  discarded at store)
    const int nc = imin(n, N2 - 1);             // clamped W2 row
    const float* __restrict__ pB = W2 + nc * N1;
    v8f acc = vzero8();
#pragma unroll 5
    for (int ks = 0; ks < 25; ++ks) {
      const int k = ks * 4 + khi;
      v2f a, bb;
      a[0] = T2[m][k];
      a[1] = T2[m][k + 1];
      bb[0] = pB[k];
      bb[1] = pB[k + 1];
      acc = wmma4(a, bb, acc);
    }
#pragma unroll
    for (int r = 0; r < 8; ++r) {
      const int o = ot * 16 + r + 8 * hi;
      if (o < N2 && n < N2)
        Z[(size_t)b * (N2 * N2) + o * N2 + n] = sinhf(acc[r]);
    }
  }
}

// ---------------------------------------------------------------------------
// Kernel B: logeig of 25x25 symmetric via cyclic Jacobi (10 fixed sweeps).
// One wave32 per batch; barriers in single-wave workgroups are near-free.
// out = V diag(log(max(d, 1e-6))) V^T  (invariant to eigen order/sign).
// ---------------------------------------------------------------------------
__global__ __launch_bounds__(32) void spd_logeig_kernel(
    const float* __restrict__ Z, float* __restrict__ out) {
  __shared__ float Am[25][27];
  __shared__ float Vm[25][27];
  __shared__ float lv[25];
  const int b = blockIdx.x;
  const int l = threadIdx.x;
  const float* Zb = Z + (size_t)b * 625;

  for (int idx = l; idx < 625; idx += 32) {
    const int i = idx / 25;
    const int j = idx - i * 25;
    Am[i][j] = Zb[idx];
    Vm[i][j] = (i == j) ? 1.0f : 0.0f;
  }
  __syncthreads();

  for (int sweep = 0; sweep < 10; ++sweep) {
    for (int p = 0; p < 24; ++p) {
      for (int q = p + 1; q < 25; ++q) {
        const float apq = Am[p][q];
        const float app = Am[p][p];
        const float aqq = Am[q][q];
        const bool rot = fabsf(apq) > 1e-20f;  // uniform across the wave
        float c = 1.0f, s = 0.0f;
        if (rot) {
          const float tau = (aqq - app) / (2.0f * apq);
          const float t = (tau >= 0.0f ? 1.0f : -1.0f) /
                          (fabsf(tau) + sqrtf(1.0f + tau * tau));
          c = 1.0f / sqrtf(1.0f + t * t);
          s = t * c;
        }
        if (rot && l < 25) {  // column rotation: A <- A*G, V <- V*G
          const float akp = Am[l][p], akq = Am[l][q];
          Am[l][p] = c * akp - s * akq;
          Am[l][q] = s * akp + c * akq;
          const float vkp = Vm[l][p], vkq = Vm[l][q];
          Vm[l][p] = c * vkp - s * vkq;
          Vm[l][q] = s * vkp + c * vkq;
        }
        __syncthreads();
        if (rot && l < 25) {  // row rotation: A <- G^T * A
          const float apk = Am[p][l], aqk = Am[q][l];
          Am[p][l] = c * apk - s * aqk;
          Am[q][l] = s * apk + c * aqk;
        }
        __syncthreads();
      }
    }
  }

  if (l < 25) lv[l] = logf(fmaxf(Am[l][l], 1e-6f));
  __syncthreads();
  if (l < 25) {
    for (int i = 0; i < 25; ++i) {
      float acc = 0.0f;
      for (int k = 0; k < 25; ++k) acc = fmaf(Vm[i][k] * lv[k], Vm[l][k], acc);
      out[(size_t)b * 625 + i * 25 + l] = acc;
    }
  }
}

extern "C" void kernel_launch(void* const* d_in, const int* in_sizes, int n_in,
                              void* d_out, int out_size, void* d_ws,
                              size_t ws_size, hipStream_t stream) {
  (void)n_in;
  (void)out_size;
  (void)ws_size;
  const float* x = (const float*)d_in[0];
  const float* W1 = (const float*)d_in[1];
  const float* W2 = (const float*)d_in[2];
  float* out = (float*)d_out;
  float* Zws = (float*)d_ws;  // B * 625 floats = 5.12 MB scratch
  const int B = in_sizes[0] / (N0 * N0);

  spd_bimap_kernel<<<B, 256, 0, stream>>>(x, W1, W2, Zws);
  spd_logeig_kernel<<<B, 32, 0, stream>>>(Zws, out);
}